// KinematicLayer_45921790328946
// MI455X (gfx1250) — compile-verified
//
#include <hip/hip_runtime.h>
#include <hip/hip_bf16.h>

// Kinematic bicycle-model scan: B=262144 trajectories, T=128 steps.
// Bandwidth-bound (~0.8 GB traffic -> ~35us @ 23.3 TB/s).
// CDNA5 path: per-wave double-buffered GLOBAL_LOAD_ASYNC_TO_LDS_B128 staging
// of motion params (ASYNCcnt pipelined, no barriers needed since each wave
// consumes only its own async loads, which complete in order).

#define T_STEPS   128
#define TC        8                    // timesteps per staged chunk
#define NCH       (T_STEPS / TC)       // 16 chunks
#define ROW_DW    (TC * 2 + 4)         // 20 dwords (80B) per trajectory row, padded, 16B aligned
#define BUF_DW    (256 * ROW_DW)       // one buffer: 20 KB
#define NTHREADS  256

__device__ __forceinline__ float clampf(float v, float lo, float hi) {
    return fminf(fmaxf(v, lo), hi);
}

#define K_DT      0.04f
#define K_DT2     0.0008f              // DT*DT/2
#define K_AX      9.0f
#define K_PSID    1.244f
#define K_SPD     20.0f

__device__ __forceinline__ void kin_step(float m0, float m1,
                                         float& x, float& y, float& speed, float& psi) {
    float ax = clampf(m0, -K_AX, K_AX);
    float pd = clampf(m1, -K_PSID, K_PSID);
    float c = __cosf(psi);
    float s = __sinf(psi);
    float sc = speed * c;
    float ss = speed * s;
    x = x + sc * K_DT + (ax * c - pd * ss) * K_DT2;
    y = y + ss * K_DT + (ax * s + pd * sc) * K_DT2;
    speed = clampf(speed + ax * K_DT, -K_SPD, K_SPD);
    psi = psi + pd * K_DT;
}

__global__ __launch_bounds__(NTHREADS) void kin_scan_async(
    const float* __restrict__ motion,   // (B, T, 2)
    const float* __restrict__ init,     // (B, 4)
    float*       __restrict__ out)      // (B, T, 4)
{
    __shared__ float smem[2 * BUF_DW];  // 40 KB: double-buffered motion tiles

    const int tid  = threadIdx.x;
    const int wave = tid >> 5;
    const int lane = tid & 31;
    const long b   = (long)blockIdx.x * NTHREADS + tid;
    const int b0w  = blockIdx.x * NTHREADS + (wave << 5);   // first trajectory of this wave

    // Async-copy lane mapping: per instruction j (0..3), 32 lanes cover
    // 8 trajectory rows x 4 sixteen-byte chunks (64B = 8 timesteps of motion).
    const int r0  = lane >> 2;          // row sub-index 0..7
    const int c16 = (lane & 3) << 4;    // byte chunk 0/16/32/48

    unsigned voff_base[4];              // global byte offset (GVS: SGPR base + VGPR i32 offset)
    unsigned lds_base[4];               // LDS byte offset within workgroup allocation
    const unsigned smem_off = (unsigned)(unsigned long long)(&smem[0]);
    #pragma unroll
    for (int j = 0; j < 4; ++j) {
        int row = (j << 3) + r0;                                     // 0..31 within wave
        voff_base[j] = (unsigned)((b0w + row) * (T_STEPS * 2 * 4)) + (unsigned)c16;
        lds_base[j]  = smem_off
                     + (unsigned)((((wave << 5) + row) * (ROW_DW * 4))) + (unsigned)c16;
    }

    // Issue one chunk's 4 async b128 copies for this wave (2KB: 32 rows x 64B).
    auto issue_chunk = [&](int k) {
        const unsigned gk = (unsigned)(k * (TC * 2 * 4));            // 64B per row per chunk
        const unsigned lk = (unsigned)((k & 1) * (BUF_DW * 4));
        #pragma unroll
        for (int j = 0; j < 4; ++j) {
            asm volatile("global_load_async_to_lds_b128 %0, %1, %2"
                         :: "v"(lds_base[j] + lk), "v"(voff_base[j] + gk), "s"(motion)
                         : "memory");
        }
    };

    float4 st = ((const float4*)init)[b];
    float x = st.x, y = st.y, speed = st.z, psi = st.w;
    float4* outp = ((float4*)out) + b * T_STEPS;

    issue_chunk(0);

    for (int k = 0; k < NCH; ++k) {
        if (k + 1 < NCH) {
            issue_chunk(k + 1);
            // 8 outstanding; oldest 4 (chunk k) done when ASYNCcnt <= 4 (in-order).
            asm volatile("s_wait_asynccnt 0x4" ::: "memory");
        } else {
            asm volatile("s_wait_asynccnt 0x0" ::: "memory");
        }
        const float* rowp = &smem[(k & 1) * BUF_DW + tid * ROW_DW];
        #pragma unroll
        for (int t = 0; t < TC; ++t) {
            float2 m = *(const float2*)&rowp[2 * t];
            kin_step(m.x, m.y, x, y, speed, psi);
            outp[k * TC + t] = make_float4(x, y, speed, psi);
        }
    }
}

// Scalar fallback for any B remainder not divisible by 256 (not expected for B=262144).
__global__ void kin_scan_tail(const float* __restrict__ motion,
                              const float* __restrict__ init,
                              float* __restrict__ out,
                              long bstart, long btotal)
{
    long b = bstart + (long)blockIdx.x * blockDim.x + threadIdx.x;
    if (b >= btotal) return;
    float4 st = ((const float4*)init)[b];
    float x = st.x, y = st.y, speed = st.z, psi = st.w;
    const float2* mp = ((const float2*)motion) + b * T_STEPS;
    float4* outp = ((float4*)out) + b * T_STEPS;
    for (int t = 0; t < T_STEPS; ++t) {
        float2 m = mp[t];
        kin_step(m.x, m.y, x, y, speed, psi);
        outp[t] = make_float4(x, y, speed, psi);
    }
}

extern "C" void kernel_launch(void* const* d_in, const int* in_sizes, int n_in,
                              void* d_out, int out_size, void* d_ws, size_t ws_size,
                              hipStream_t stream) {
    const float* motion = (const float*)d_in[0];   // (B, T, 2) f32
    const float* init   = (const float*)d_in[1];   // (B, 4)    f32
    float* out          = (float*)d_out;           // (B, T, 4) f32

    const long B     = (long)in_sizes[1] / 4;
    const long Bmain = B & ~(long)(NTHREADS - 1);

    if (Bmain > 0) {
        kin_scan_async<<<(int)(Bmain / NTHREADS), NTHREADS, 0, stream>>>(motion, init, out);
    }
    if (B > Bmain) {
        kin_scan_tail<<<1, NTHREADS, 0, stream>>>(motion, init, out, Bmain, B);
    }
}